// ROISetAbstraction_30253749633643
// MI455X (gfx1250) — compile-verified
//
#include <hip/hip_runtime.h>
#include <hip/hip_bf16.h>

typedef float v2f __attribute__((ext_vector_type(2)));
typedef float v8f __attribute__((ext_vector_type(8)));

namespace {
constexpr int BATCH = 2;
constexpr int KPTS  = 2048;
constexpr int NRAW  = 16384;
constexpr int NVOX  = 8192;
constexpr int FEAT_DIM = 416;   // 256 bev + 16 + 16 + 64 + 64
constexpr int OUT_DIM  = 128;
constexpr int BEV_C = 256, BEV_H = 200, BEV_W = 176;
constexpr float VOXSZ = 0.05f, STRIDEF = 8.0f;
}

// ---------------------------------------------------------------------------
// Ball query: one wave32 per keypoint. In-order compaction via ballot gives
// the first `ns` in-radius indices in original order (matches CUDA
// ball_query / the reference's top_k(-order) trick), slots past cnt get the
// first index, empty balls get cnt=0 and idx filled with 0.
// ---------------------------------------------------------------------------
__global__ void __launch_bounds__(256)
ball_query_kernel(const float* __restrict__ xyz,     // (B,N,3)
                  const float* __restrict__ newxyz,  // (B,K,3)
                  int* __restrict__ idx,             // (B,K,ns)
                  int* __restrict__ cnt,             // (B,K)
                  int N, int ns, float r2) {
  int gid  = blockIdx.x * blockDim.x + threadIdx.x;
  int wid  = gid >> 5;
  int lane = threadIdx.x & 31;
  if (wid >= BATCH * KPTS) return;
  int b = wid / KPTS;
  const float* nb = newxyz + (size_t)wid * 3;
  float cx = nb[0], cy = nb[1], cz = nb[2];
  const float* xb = xyz + (size_t)b * N * 3;
  int* out = idx + (size_t)wid * ns;

  int found = 0;
  int first_i = 0x7fffffff;
  for (int base = 0; base < N && found < ns; base += 32) {
    int i = base + lane;
    bool valid = false;
    if (i < N) {
      float dx = xb[i*3+0] - cx;
      float dy = xb[i*3+1] - cy;
      float dz = xb[i*3+2] - cz;
      valid = (dx*dx + dy*dy + dz*dz) < r2;
    }
    unsigned mask = (unsigned)__ballot(valid);   // wave32: low 32 bits
    int pos = found + __popc(mask & ((1u << lane) - 1u));
    if (valid && pos < ns) {
      out[pos] = i;
      if (pos == 0) first_i = i;
    }
    found += __popc(mask);
  }
  if (lane == 0) cnt[wid] = found;
  // broadcast the first in-radius index across the wave
  for (int o = 16; o > 0; o >>= 1) {
    int other = __shfl_xor(first_i, o, 32);
    first_i = first_i < other ? first_i : other;
  }
  int stored = found < ns ? found : ns;
  int fill = (stored > 0) ? first_i : 0;
  for (int s = stored + lane; s < ns; s += 32) out[s] = fill;
}

// ---------------------------------------------------------------------------
// BEV bilinear interpolation; one block per keypoint, one thread per channel.
// Clipped corner coordinates are used in the weights (matches reference).
// ---------------------------------------------------------------------------
__global__ void __launch_bounds__(BEV_C)
bev_interp_kernel(const float* __restrict__ kp,
                  const float* __restrict__ bev,   // (B,256,H,W)
                  float* __restrict__ feats) {     // (B*K, FEAT_DIM)
  int pk = blockIdx.x;            // 0 .. B*K-1
  int c  = threadIdx.x;           // 0 .. 255
  int b  = pk / KPTS;
  float px = kp[pk*3 + 0], py = kp[pk*3 + 1];
  float xi = px / VOXSZ / STRIDEF;
  float yi = (py + 40.0f) / VOXSZ / STRIDEF;
  int x0 = (int)floorf(xi), y0 = (int)floorf(yi);
  int x1 = x0 + 1, y1 = y0 + 1;
  int x0c = min(max(x0, 0), BEV_W - 1), x1c = min(max(x1, 0), BEV_W - 1);
  int y0c = min(max(y0, 0), BEV_H - 1), y1c = min(max(y1, 0), BEV_H - 1);
  float x0f = (float)x0c, x1f = (float)x1c, y0f = (float)y0c, y1f = (float)y1c;
  float wa = (x1f - xi) * (y1f - yi);
  float wb = (x1f - xi) * (yi  - y0f);
  float wc = (xi  - x0f) * (y1f - yi);
  float wd = (xi  - x0f) * (yi  - y0f);
  const float* im = bev + ((size_t)b * BEV_C + c) * (size_t)(BEV_H * BEV_W);
  float Ia = im[y0c*BEV_W + x0c], Ib = im[y1c*BEV_W + x0c];
  float Ic = im[y0c*BEV_W + x1c], Id = im[y1c*BEV_W + x1c];
  feats[(size_t)pk * FEAT_DIM + c] = Ia*wa + Ib*wb + Ic*wc + Id*wd;
}

// grouped-input element: col 0..2 = relative xyz, col 3..3+C-1 = point feats,
// beyond = zero padding; entire row zero for empty balls.
template<int C>
__device__ __forceinline__ float gelem(bool v, float rx, float ry, float rz,
                                       const float* __restrict__ fp, int col) {
  if (!v) return 0.0f;
  if (col == 0) return rx;
  if (col == 1) return ry;
  if (col == 2) return rz;
  int c = col - 3;
  if (c < C) return fp[c];
  return 0.0f;
}

// ---------------------------------------------------------------------------
// Layer-1 GEMM with on-the-fly gather:  H1 = relu((G @ W) * gamma + beta)
// G rows are (keypoint, neighbor) slots; one wave computes a 16-row x CMID
// tile with V_WMMA_F32_16X16X4_F32, K stepped by 4 (CIN padded to mult of 4).
// A frag (16x4 f32): lanes 0-15 hold K = 4k+{0,1}, lanes 16-31 hold K = 4k+{2,3}.
// ---------------------------------------------------------------------------
template<int C, int CMID, int NS>
__global__ void __launch_bounds__(256)
gemm_gather_wmma(const float* __restrict__ src_xyz,
                 const float* __restrict__ src_feats, int N,
                 const float* __restrict__ keypts,
                 const int* __restrict__ idx, const int* __restrict__ cnt,
                 const float* __restrict__ Wt,     // (3+C, CMID) row-major
                 const float* __restrict__ gamma, const float* __restrict__ beta,
                 float* __restrict__ H1) {         // (R, CMID)
  constexpr int CIN = 3 + C;
  constexpr int NT = CMID / 16;
  constexpr int KSTEPS = (CIN + 3) / 4;
  constexpr int R = BATCH * KPTS * NS;
  int gid  = blockIdx.x * blockDim.x + threadIdx.x;
  int wid  = gid >> 5;
  int lane = threadIdx.x & 31;
  int row0 = wid * 16;
  if (row0 >= R) return;                    // wave-uniform, EXEC stays full

  int m   = lane & 15;
  int r   = row0 + m;
  int key = r / NS;                         // = b*K + k
  int s   = r % NS;
  int b   = key / KPTS;
  bool v  = cnt[key] > 0;                   // empty_ball_mask
  int  j  = idx[(size_t)key * NS + s];
  const float* kp = keypts + (size_t)key * 3;
  const float* sx = src_xyz + ((size_t)b * N + j) * 3;
  float rx = sx[0] - kp[0], ry = sx[1] - kp[1], rz = sx[2] - kp[2];
  const float* fp = src_feats + ((size_t)b * N + j) * C;

  int khalf = (lane >> 4) << 1;             // 0 for lanes 0-15, 2 for 16-31
  int ncol  = lane & 15;

  v8f acc[NT];
  #pragma unroll
  for (int t = 0; t < NT; ++t)
    #pragma unroll
    for (int e = 0; e < 8; ++e) acc[t][e] = 0.0f;

  #pragma unroll
  for (int kk = 0; kk < KSTEPS; ++kk) {
    int k0 = kk * 4 + khalf;
    v2f a;
    a.x = gelem<C>(v, rx, ry, rz, fp, k0);
    a.y = gelem<C>(v, rx, ry, rz, fp, k0 + 1);
    #pragma unroll
    for (int t = 0; t < NT; ++t) {
      int n = t * 16 + ncol;
      v2f bf;
      bf.x = (k0     < CIN) ? Wt[(size_t)k0       * CMID + n] : 0.0f;
      bf.y = (k0 + 1 < CIN) ? Wt[(size_t)(k0 + 1) * CMID + n] : 0.0f;
      acc[t] = __builtin_amdgcn_wmma_f32_16x16x4_f32(
          false, a, false, bf, (short)0, acc[t], false, false);
    }
  }
  // D layout: VGPR i -> row (i + 8*(lane>=16)), col = lane&15 within tile
  int mh = (lane >> 4) * 8;
  #pragma unroll
  for (int t = 0; t < NT; ++t) {
    int n = t * 16 + ncol;
    float ga = gamma[n], be = beta[n];
    #pragma unroll
    for (int i = 0; i < 8; ++i) {
      int rr = row0 + i + mh;
      float o = acc[t][i] * ga + be;
      o = o > 0.0f ? o : 0.0f;
      H1[(size_t)rr * CMID + n] = o;
    }
  }
}

// ---------------------------------------------------------------------------
// Dense GEMM: out = relu((A @ W) * gamma + beta).
// ATOMIC=true: fuse neighbor max-pool via integer atomicMax (values are >= 0
// post-ReLU) into feats[key, colOff + n].  ATOMIC=false: direct store (fusion).
// ---------------------------------------------------------------------------
template<int CIN, int COUT, bool ATOMIC>
__global__ void __launch_bounds__(256)
gemm_dense_wmma(const float* __restrict__ A,      // (R, CIN)
                const float* __restrict__ Wt,     // (CIN, COUT)
                const float* __restrict__ gamma, const float* __restrict__ beta,
                float* __restrict__ out, int R, int ldo, int colOff, int ns) {
  constexpr int NT = COUT / 16;
  constexpr int KSTEPS = (CIN + 3) / 4;
  int gid  = blockIdx.x * blockDim.x + threadIdx.x;
  int wid  = gid >> 5;
  int lane = threadIdx.x & 31;
  int row0 = wid * 16;
  if (row0 >= R) return;

  int r = row0 + (lane & 15);
  const float* arow = A + (size_t)r * CIN;
  int khalf = (lane >> 4) << 1;
  int ncol  = lane & 15;

  v8f acc[NT];
  #pragma unroll
  for (int t = 0; t < NT; ++t)
    #pragma unroll
    for (int e = 0; e < 8; ++e) acc[t][e] = 0.0f;

  for (int kk = 0; kk < KSTEPS; ++kk) {
    int k0 = kk * 4 + khalf;
    v2f a;
    a.x = (k0     < CIN) ? arow[k0]     : 0.0f;
    a.y = (k0 + 1 < CIN) ? arow[k0 + 1] : 0.0f;
    #pragma unroll
    for (int t = 0; t < NT; ++t) {
      int n = t * 16 + ncol;
      v2f bf;
      bf.x = (k0     < CIN) ? Wt[(size_t)k0       * COUT + n] : 0.0f;
      bf.y = (k0 + 1 < CIN) ? Wt[(size_t)(k0 + 1) * COUT + n] : 0.0f;
      acc[t] = __builtin_amdgcn_wmma_f32_16x16x4_f32(
          false, a, false, bf, (short)0, acc[t], false, false);
    }
  }
  int mh = (lane >> 4) * 8;
  #pragma unroll
  for (int t = 0; t < NT; ++t) {
    int n = t * 16 + ncol;
    float ga = gamma[n], be = beta[n];
    #pragma unroll
    for (int i = 0; i < 8; ++i) {
      int rr = row0 + i + mh;
      float o = acc[t][i] * ga + be;
      o = o > 0.0f ? o : 0.0f;
      if (ATOMIC) {
        int key = rr / ns;  // rows are (key, neighbor)-contiguous
        atomicMax((int*)(out + (size_t)key * ldo + colOff + n),
                  __float_as_int(o));
      } else {
        out[(size_t)rr * ldo + colOff + n] = o;
      }
    }
  }
}

// ---------------------------------------------------------------------------
extern "C" void kernel_launch(void* const* d_in, const int* in_sizes, int n_in,
                              void* d_out, int out_size, void* d_ws, size_t ws_size,
                              hipStream_t stream) {
  const float* keypoints = (const float*)d_in[0];
  const float* bev       = (const float*)d_in[1];
  const float* raw_xyz   = (const float*)d_in[2];
  const float* raw_feats = (const float*)d_in[3];
  const float* c4_xyz    = (const float*)d_in[4];
  const float* c4_feats  = (const float*)d_in[5];
  const float* rW00=(const float*)d_in[6],  *rW01=(const float*)d_in[7];
  const float* rg00=(const float*)d_in[8],  *rb00=(const float*)d_in[9];
  const float* rg01=(const float*)d_in[10], *rb01=(const float*)d_in[11];
  const float* rW10=(const float*)d_in[12], *rW11=(const float*)d_in[13];
  const float* rg10=(const float*)d_in[14], *rb10=(const float*)d_in[15];
  const float* rg11=(const float*)d_in[16], *rb11=(const float*)d_in[17];
  const float* cW00=(const float*)d_in[18], *cW01=(const float*)d_in[19];
  const float* cg00=(const float*)d_in[20], *cb00=(const float*)d_in[21];
  const float* cg01=(const float*)d_in[22], *cb01=(const float*)d_in[23];
  const float* cW10=(const float*)d_in[24], *cW11=(const float*)d_in[25];
  const float* cg10=(const float*)d_in[26], *cb10=(const float*)d_in[27];
  const float* cg11=(const float*)d_in[28], *cb11=(const float*)d_in[29];
  const float* fW =(const float*)d_in[30];
  const float* fg =(const float*)d_in[31], *fb=(const float*)d_in[32];

  // workspace carve-up (~42 MB total)
  char* ws = (char*)d_ws;
  size_t off = 0;
  auto carve = [&](size_t bytes) -> char* {
    char* p = ws + off;
    off = (off + bytes + 255) & ~(size_t)255;
    return p;
  };
  const size_t feats_bytes = (size_t)BATCH * KPTS * FEAT_DIM * sizeof(float);
  float* feats  = (float*)carve(feats_bytes);
  int*   idx_r0 = (int*)carve((size_t)BATCH * KPTS * 16 * sizeof(int));
  int*   idx_r1 = (int*)carve((size_t)BATCH * KPTS * 32 * sizeof(int));
  int*   idx_c0 = (int*)carve((size_t)BATCH * KPTS * 16 * sizeof(int));
  int*   idx_c1 = (int*)carve((size_t)BATCH * KPTS * 32 * sizeof(int));
  int*   cnt_r0 = (int*)carve((size_t)BATCH * KPTS * sizeof(int));
  int*   cnt_r1 = (int*)carve((size_t)BATCH * KPTS * sizeof(int));
  int*   cnt_c0 = (int*)carve((size_t)BATCH * KPTS * sizeof(int));
  int*   cnt_c1 = (int*)carve((size_t)BATCH * KPTS * sizeof(int));
  float* H1     = (float*)carve((size_t)BATCH * KPTS * 32 * 64 * sizeof(float));

  // feats must start at 0: SA columns are accumulated with atomic max.
  hipMemsetAsync(feats, 0, feats_bytes, stream);

  // BEV bilinear -> feats[:, 0:256]
  bev_interp_kernel<<<BATCH * KPTS, BEV_C, 0, stream>>>(keypoints, bev, feats);

  // ball queries (one wave per keypoint)
  int bqBlocks = (BATCH * KPTS * 32 + 255) / 256;
  ball_query_kernel<<<bqBlocks, 256, 0, stream>>>(raw_xyz, keypoints, idx_r0, cnt_r0, NRAW, 16, 0.8f*0.8f);
  ball_query_kernel<<<bqBlocks, 256, 0, stream>>>(raw_xyz, keypoints, idx_r1, cnt_r1, NRAW, 32, 1.2f*1.2f);
  ball_query_kernel<<<bqBlocks, 256, 0, stream>>>(c4_xyz,  keypoints, idx_c0, cnt_c0, NVOX, 16, 2.4f*2.4f);
  ball_query_kernel<<<bqBlocks, 256, 0, stream>>>(c4_xyz,  keypoints, idx_c1, cnt_c1, NVOX, 32, 4.8f*4.8f);

  auto blocksFor = [](int R) { return ((R / 16) * 32 + 255) / 256; };

  // raw group 0: ns=16, CIN=4, CMID=16 -> feats[:, 256:272]
  {
    int R = BATCH * KPTS * 16;
    gemm_gather_wmma<1, 16, 16><<<blocksFor(R), 256, 0, stream>>>(
        raw_xyz, raw_feats, NRAW, keypoints, idx_r0, cnt_r0, rW00, rg00, rb00, H1);
    gemm_dense_wmma<16, 16, true><<<blocksFor(R), 256, 0, stream>>>(
        H1, rW01, rg01, rb01, feats, R, FEAT_DIM, 256, 16);
  }
  // raw group 1: ns=32 -> feats[:, 272:288]
  {
    int R = BATCH * KPTS * 32;
    gemm_gather_wmma<1, 16, 32><<<blocksFor(R), 256, 0, stream>>>(
        raw_xyz, raw_feats, NRAW, keypoints, idx_r1, cnt_r1, rW10, rg10, rb10, H1);
    gemm_dense_wmma<16, 16, true><<<blocksFor(R), 256, 0, stream>>>(
        H1, rW11, rg11, rb11, feats, R, FEAT_DIM, 272, 32);
  }
  // c4 group 0: ns=16, CIN=67, CMID=64 -> feats[:, 288:352]
  {
    int R = BATCH * KPTS * 16;
    gemm_gather_wmma<64, 64, 16><<<blocksFor(R), 256, 0, stream>>>(
        c4_xyz, c4_feats, NVOX, keypoints, idx_c0, cnt_c0, cW00, cg00, cb00, H1);
    gemm_dense_wmma<64, 64, true><<<blocksFor(R), 256, 0, stream>>>(
        H1, cW01, cg01, cb01, feats, R, FEAT_DIM, 288, 16);
  }
  // c4 group 1: ns=32 -> feats[:, 352:416]
  {
    int R = BATCH * KPTS * 32;
    gemm_gather_wmma<64, 64, 32><<<blocksFor(R), 256, 0, stream>>>(
        c4_xyz, c4_feats, NVOX, keypoints, idx_c1, cnt_c1, cW10, cg10, cb10, H1);
    gemm_dense_wmma<64, 64, true><<<blocksFor(R), 256, 0, stream>>>(
        H1, cW11, cg11, cb11, feats, R, FEAT_DIM, 352, 32);
  }
  // fusion: (4096,416) @ (416,128), relu(h*fg+fb) -> d_out
  {
    int R = BATCH * KPTS;
    gemm_dense_wmma<416, 128, false><<<blocksFor(R), 256, 0, stream>>>(
        feats, fW, fg, fb, (float*)d_out, R, OUT_DIM, 0, 1);
  }
}